// Net_TDE_48180943127013
// MI455X (gfx1250) — compile-verified
//
#include <hip/hip_runtime.h>
#include <stdint.h>

// Problem constants (from reference): T=2048, B=8192, 2 channels.
#define T_STEPS   2048
#define BATCH     8192
#define B2        (BATCH * 2)          // floats per timestep row of x / cur
#define CT        16                   // timesteps staged per TDM tile
#define NCHUNK    (T_STEPS / CT)       // 128 TDM ops per wave
#define WAVES_PB  2
#define BLOCK_THR (WAVES_PB * 32)
#define TILE_F32  (CT * 32 * 2)        // 1024 floats = 4 KB per buffer

typedef unsigned int u32x4 __attribute__((ext_vector_type(4)));
typedef int          i32x8 __attribute__((ext_vector_type(8)));
typedef int          i32x4 __attribute__((ext_vector_type(4)));

// Issue one TDM load: 2D tile (CT rows x 64 f32), row stride B2 elements,
// packed contiguously into LDS at lds_off. Descriptor bit layout per
// CDNA5 ISA (group0 128b, group1 256b).
__device__ __forceinline__ void tdm_load_tile(uint32_t lds_off, uint64_t gaddr)
{
    u32x4 g0;
    g0[0] = 1u;                                  // count=1 (valid user D#)
    g0[1] = lds_off;                             // lds_addr (bytes)
    g0[2] = (uint32_t)gaddr;                     // global_addr[31:0]
    g0[3] = (uint32_t)(gaddr >> 32) | (2u << 30);// global_addr[56:32] | type=2

    i32x8 g1;
    g1[0] = (int)(2u << 16);                     // data_size=2 (4 bytes)
    g1[1] = (int)(((uint32_t)B2 & 0xFFFFu) << 16);          // tensor_dim0 lo16
    g1[2] = (int)(((uint32_t)B2 >> 16) |
                  (((uint32_t)T_STEPS & 0xFFFFu) << 16));   // td0 hi16 | td1 lo16
    g1[3] = (int)(((uint32_t)T_STEPS >> 16) | (64u << 16)); // td1 hi16 | tile_dim0=64
    g1[4] = (int)((uint32_t)CT);                 // tile_dim1=CT, tile_dim2=0
    g1[5] = (int)((uint32_t)B2);                 // tensor_dim0_stride lo32
    g1[6] = 0;                                   // stride hi16 | dim1_stride lo16
    g1[7] = 0;                                   // dim1_stride hi32

    i32x4 z4 = {0, 0, 0, 0};                     // groups 2/3 unused (2D tensor)
    i32x8 z8 = {0, 0, 0, 0, 0, 0, 0, 0};         // extra operand (clang-23 form)
    __builtin_amdgcn_tensor_load_to_lds(g0, g1, z4, z4, z8, 0);
}

__global__ __launch_bounds__(BLOCK_THR) void
tde_scan_kernel(const float* __restrict__ x, const float* __restrict__ W,
                float* __restrict__ out)
{
    __shared__ __align__(16) float smem[WAVES_PB * 2 * TILE_F32];

    const int lane   = threadIdx.x & 31;
    const int waveId = threadIdx.x >> 5;
    const int batch0 = blockIdx.x * BLOCK_THR + waveId * 32; // wave's batch start
    const int b      = batch0 + lane;

    float* wbuf = &smem[waveId * 2 * TILE_F32];
    const uint32_t lds0 = (uint32_t)(uintptr_t)(void*)wbuf;  // low 32b of flat = LDS offset
    const uint32_t ldsB[2] = { lds0, lds0 + TILE_F32 * 4u };

    const float w00 = W[0], w01 = W[1], w10 = W[2], w11 = W[3];

    float* out_cur  = out;                                       // (T, B, 2)
    float* out_spk  = out + (size_t)T_STEPS * B2;                // (T, B)
    float* out_gain = out_spk  + (size_t)T_STEPS * BATCH;
    float* out_epsc = out_gain + (size_t)T_STEPS * BATCH;
    float* out_mem  = out_epsc + (size_t)T_STEPS * BATCH;

    const uint64_t gbase = (uint64_t)(uintptr_t)x + 8ull * (uint64_t)batch0;
    const uint64_t chunkStride = 4ull * (uint64_t)CT * (uint64_t)B2; // bytes / CT rows

    // Prime the double buffer.
    tdm_load_tile(ldsB[0], gbase);
    tdm_load_tile(ldsB[1], gbase + chunkStride);

    float gain = 0.0f, epsc = 0.0f, mem = 0.0f;

    for (int c = 0; c < NCHUNK; ++c) {
        // Oldest outstanding TDM op holds chunk c; allow the newer one to fly.
        if (c + 1 < NCHUNK) __builtin_amdgcn_s_wait_tensorcnt(1);
        else                __builtin_amdgcn_s_wait_tensorcnt(0);

        const float2* buf =
            (const float2*)&smem[waveId * 2 * TILE_F32 + (c & 1) * TILE_F32];
        const int tbase = c * CT;

        #pragma unroll
        for (int tt = 0; tt < CT; ++tt) {
            float2 xv = buf[tt * 32 + lane];               // ds_load_b64, conflict-free

            float cur0 = fmaf(w01, xv.y, w00 * xv.x);      // x @ W^T
            float cur1 = fmaf(w11, xv.y, w10 * xv.x);

            gain = fmaf(0.96f, gain, cur0);                // A_GAIN*gain + fac
            epsc = fmaf(0.92f, epsc, gain * cur1);         // A_EPSC*epsc + gain*trig
            mem  = fmaf(0.90f, mem,  epsc);                // B_MEM*mem + epsc
            float spk = (mem > 1.0f) ? 1.0f : 0.0f;        // THR = 1.0
            mem -= spk;

            const int    t  = tbase + tt;
            const size_t r1 = (size_t)t * BATCH + (size_t)b;

            ((float2*)out_cur)[r1] = make_float2(cur0, cur1); // b64, coalesced
            out_spk [r1] = spk;
            out_gain[r1] = gain;
            out_epsc[r1] = epsc;
            out_mem [r1] = mem;
        }

        // Ensure all ds_loads from this buffer have landed before the DMA
        // engine is allowed to overwrite it (WAR on LDS), then refill.
        if (c + 2 < NCHUNK) {
            asm volatile("s_wait_dscnt 0" ::: "memory");
            tdm_load_tile(ldsB[c & 1], gbase + (uint64_t)(c + 2) * chunkStride);
        }
    }
}

extern "C" void kernel_launch(void* const* d_in, const int* in_sizes, int n_in,
                              void* d_out, int out_size, void* d_ws, size_t ws_size,
                              hipStream_t stream) {
    (void)in_sizes; (void)n_in; (void)out_size; (void)d_ws; (void)ws_size;
    const float* x = (const float*)d_in[0];   // (T, B, 2) f32
    const float* W = (const float*)d_in[1];   // (2, 2) f32
    float* out = (float*)d_out;               // cur | spks | gain | epsc | mem

    dim3 grid(BATCH / BLOCK_THR);             // 128 blocks
    dim3 block(BLOCK_THR);                    // 64 threads = 2 wave32
    tde_scan_kernel<<<grid, block, 0, stream>>>(x, W, out);
}